// GATLayer_39049842655813
// MI455X (gfx1250) — compile-verified
//
#include <hip/hip_runtime.h>
#include <stdint.h>

typedef __attribute__((ext_vector_type(2))) float v2f;
typedef __attribute__((ext_vector_type(8))) float v8f;

// GCC-vector int4 with explicit address spaces for the async-copy builtin:
// expected param type is "__attribute__((vector_size(16))) int __device__ *"
typedef int v4i_vs __attribute__((vector_size(16)));
typedef __attribute__((address_space(1))) v4i_vs* as1_v4i_p;   // global src
typedef __attribute__((address_space(3))) v4i_vs* as3_v4i_p;   // LDS dst (32-bit)

#define DIN   256
#define DOUT  128
#define NHEAD 8
#define DH    16
#define XSTRIDE 260   // 16x256 fp32 tile padded: 260 % 64 = 4 -> conflict-free column reads
                      // row stride = 1040 B (16B-aligned, ok for b128 async stores)

#if defined(__gfx1250__) && __has_builtin(__builtin_amdgcn_global_load_async_to_lds_b128)
#define GAT_ASYNC 1
#else
#define GAT_ASYNC 0
#endif

// ---------------------------------------------------------------------------
// h[N,128] = x[N,256] @ W[128,256]^T  via V_WMMA_F32_16X16X4_F32
// block = 256 threads = 8 waves; wave w computes output columns [16w, 16w+16)
// for the 16-row tile owned by the block. x-tile staged to LDS via
// GLOBAL_LOAD_ASYNC_TO_LDS_B128 when available.
// ---------------------------------------------------------------------------
__global__ __launch_bounds__(256) void gat_gemm(const float* __restrict__ x,
                                                const float* __restrict__ W,
                                                float* __restrict__ h, int N) {
    __shared__ __align__(16) float xs[16 * XSTRIDE];
    const int row0 = blockIdx.x * 16;
    const int tid  = threadIdx.x;

    // stage 16x256 x-tile into LDS: 1024 float4s, 4 per thread
    for (int i = tid; i < 16 * (DIN / 4); i += 256) {
        const int r  = i / (DIN / 4);
        const int c4 = i % (DIN / 4);
        int rr = row0 + r; if (rr >= N) rr = N - 1;      // clamp (N % 16 == 0 in practice)
        const float* gsrc = x + (size_t)rr * DIN + c4 * 4;
        float*       ldst = &xs[r * XSTRIDE + c4 * 4];
#if GAT_ASYNC
        // async DMA: memory -> LDS, no VGPR round trip (ASYNCcnt-tracked)
        __builtin_amdgcn_global_load_async_to_lds_b128(
            (as1_v4i_p)(uintptr_t)gsrc,
            (as3_v4i_p)(uint32_t)(uintptr_t)ldst,   // LDS offset = low 32 bits of generic addr
            /*offset=*/0, /*cpol=*/0);
#else
        const float4 v = *(const float4*)gsrc;
        ldst[0] = v.x; ldst[1] = v.y; ldst[2] = v.z; ldst[3] = v.w;
#endif
    }
#if GAT_ASYNC
#if __has_builtin(__builtin_amdgcn_s_wait_asynccnt)
    __builtin_amdgcn_s_wait_asynccnt(0);
#else
    asm volatile("s_wait_asynccnt 0x0" ::: "memory");
#endif
#endif
    __syncthreads();

    const int wave = tid >> 5;
    const int lane = tid & 31;
    const int o0   = wave * DH;           // output-column tile base
    const int m    = lane & 15;           // A row / B col index for this lane
    const int kh   = (lane >> 4) * 2;     // lanes 16-31 carry K+2,K+3 halves

    v8f acc = {};
    const float* wrow = W + (size_t)(o0 + m) * DIN;   // B[k][n] = W[o0+n][k]

    #pragma unroll 4
    for (int k0 = 0; k0 < DIN; k0 += 4) {
        v2f a, b;
        a.x = xs[m * XSTRIDE + k0 + kh];
        a.y = xs[m * XSTRIDE + k0 + kh + 1];
        b.x = wrow[k0 + kh];
        b.y = wrow[k0 + kh + 1];
        acc = __builtin_amdgcn_wmma_f32_16x16x4_f32(
            /*neg_a=*/false, a, /*neg_b=*/false, b,
            /*c_mod=*/(short)0, acc, /*reuse_a=*/false, /*reuse_b=*/false);
    }

    // C/D layout: VGPR v -> M=v (lanes 0-15), M=v+8 (lanes 16-31); N = lane&15
    const int mbase = (lane >> 4) * 8;
    float* hout = h + (size_t)(row0 + mbase) * DOUT + o0 + m;
    if (row0 + 16 <= N) {               // wave-uniform fast path: no exec churn
        #pragma unroll
        for (int v = 0; v < 8; ++v)
            hout[(size_t)v * DOUT] = acc[v];
    } else {
        #pragma unroll
        for (int v = 0; v < 8; ++v)
            if (row0 + mbase + v < N) hout[(size_t)v * DOUT] = acc[v];
    }
}

// ---------------------------------------------------------------------------
// zero out[N*128] and denom[N*8] (both multiples of 4 floats, 16B aligned)
// ---------------------------------------------------------------------------
__global__ void gat_zero(float4* __restrict__ out, float4* __restrict__ denom, int N) {
    const size_t i = (size_t)blockIdx.x * blockDim.x + threadIdx.x;
    const float4 z = {0.f, 0.f, 0.f, 0.f};
    if (i < (size_t)N * (DOUT / 4))  out[i]   = z;
    if (i < (size_t)N * (NHEAD / 4)) denom[i] = z;
}

// ---------------------------------------------------------------------------
// per-edge exp(leaky(score)); lane = 4 dims, head = lane>>2, 4-lane butterfly
// ---------------------------------------------------------------------------
__device__ __forceinline__ float edge_exp(const float* __restrict__ h,
                                          const float* __restrict__ Watt,
                                          float batt, int src, int dst, int lane,
                                          float4* hs_out) {
    const float4 hs = ((const float4*)(h + (size_t)src * DOUT))[lane];
    const float4 hd = ((const float4*)(h + (size_t)dst * DOUT))[lane];
    const int q = (lane & 3) * 4;
    const float4 wa = *(const float4*)(Watt + q);        // first 16: h_i = h[src]
    const float4 wb = *(const float4*)(Watt + DH + q);   // last 16:  h_j = h[dst]
    float p = hs.x * wa.x + hs.y * wa.y + hs.z * wa.z + hs.w * wa.w
            + hd.x * wb.x + hd.y * wb.y + hd.z * wb.z + hd.w * wb.w;
    p += __shfl_xor(p, 1, 32);
    p += __shfl_xor(p, 2, 32);     // all 4 lanes of a head now hold the full dot
    float s = p + batt;
    s = (s >= 0.0f) ? s : 0.2f * s;
    *hs_out = hs;
    return __expf(s);
}

// ---------------------------------------------------------------------------
// pass 1 over edges: denom[dst, head] += exp
// one wave per edge; edges >= E are self-loops (src = dst = e - E)
// ---------------------------------------------------------------------------
__global__ __launch_bounds__(256) void gat_denom(const float* __restrict__ h,
                                                 const int* __restrict__ ei,
                                                 const float* __restrict__ Watt,
                                                 const float* __restrict__ batt,
                                                 float* __restrict__ denom,
                                                 int E, int Etot) {
    const int e = blockIdx.x * 8 + (threadIdx.x >> 5);
    if (e >= Etot) return;
    const int lane = threadIdx.x & 31;
    const int src = (e < E) ? ei[e]     : (e - E);
    const int dst = (e < E) ? ei[E + e] : (e - E);
    float4 hs;
    const float ex = edge_exp(h, Watt, batt[0], src, dst, lane, &hs);
    if ((lane & 3) == 0)
        atomicAdd(&denom[(size_t)dst * NHEAD + (lane >> 2)], ex);
}

// ---------------------------------------------------------------------------
// pass 2 over edges: out[dst] += h[src] * exp/denom[dst]
// ---------------------------------------------------------------------------
__global__ __launch_bounds__(256) void gat_agg(const float* __restrict__ h,
                                               const int* __restrict__ ei,
                                               const float* __restrict__ Watt,
                                               const float* __restrict__ batt,
                                               const float* __restrict__ denom,
                                               float* __restrict__ out,
                                               int E, int Etot) {
    const int e = blockIdx.x * 8 + (threadIdx.x >> 5);
    if (e >= Etot) return;
    const int lane = threadIdx.x & 31;
    const int src = (e < E) ? ei[e]     : (e - E);
    const int dst = (e < E) ? ei[E + e] : (e - E);
    float4 hs;
    const float ex  = edge_exp(h, Watt, batt[0], src, dst, lane, &hs);
    const float att = ex / denom[(size_t)dst * NHEAD + (lane >> 2)];
    float* o = out + (size_t)dst * DOUT + lane * 4;
    atomicAdd(o + 0, hs.x * att);
    atomicAdd(o + 1, hs.y * att);
    atomicAdd(o + 2, hs.z * att);
    atomicAdd(o + 3, hs.w * att);
}

extern "C" void kernel_launch(void* const* d_in, const int* in_sizes, int n_in,
                              void* d_out, int out_size, void* d_ws, size_t ws_size,
                              hipStream_t stream) {
    const float* x    = (const float*)d_in[0];   // [N, 256]
    const float* W    = (const float*)d_in[1];   // [128, 256]
    const float* Watt = (const float*)d_in[2];   // [1, 32]
    const float* batt = (const float*)d_in[3];   // [1]
    const int*   ei   = (const int*)d_in[4];     // [2, E]

    const int N    = in_sizes[0] / DIN;
    const int E    = in_sizes[4] / 2;
    const int Etot = E + N;

    float* out   = (float*)d_out;
    float* h     = (float*)d_ws;                       // N*128 floats
    float* denom = h + (size_t)N * DOUT;               // N*8 floats

    const size_t zt = (size_t)N * (DOUT / 4);
    gat_zero<<<(int)((zt + 255) / 256), 256, 0, stream>>>((float4*)out, (float4*)denom, N);

    gat_gemm<<<(N + 15) / 16, 256, 0, stream>>>(x, W, h, N);

    const int eb = (Etot + 7) / 8;
    gat_denom<<<eb, 256, 0, stream>>>(h, ei, Watt, batt, denom, E, Etot);
    gat_agg  <<<eb, 256, 0, stream>>>(h, ei, Watt, batt, denom, out, E, Etot);
}